// TEEDAM_88227218195093
// MI455X (gfx1250) — compile-verified
//
#include <hip/hip_runtime.h>
#include <math.h>

// Problem constants (from reference): B=16, P=1024, M=4, D=64
#define B_    16
#define P_    1024
#define M_    4
#define D_    64
#define NT_   64                 // P / 16 tiles
#define NTOT_ (B_ * M_ * NT_)    // 4096 (b,m,tile) triples

typedef __attribute__((ext_vector_type(2))) float v2f;
typedef __attribute__((ext_vector_type(8))) float v8f;

#if __has_builtin(__builtin_amdgcn_wmma_f32_16x16x4_f32)
#define USE_WMMA_F32X4 1
#endif

// ---------------------------------------------------------------------------
// Kernel A: per-(b,m,tile) local summary: Mx = max(pre), E = sum exp(pre-Mx),
//           S[d] = sum_q exp(pre_q - Mx) * V[q][d].  One wave per tile.
// ---------------------------------------------------------------------------
__global__ void __launch_bounds__(32)
tile_summary_kernel(const float* __restrict__ pre,
                    const float* __restrict__ vals,
                    float* __restrict__ wsMx,
                    float* __restrict__ wsE,
                    float* __restrict__ wsS) {
  int gt   = blockIdx.x;              // (b*M + m)*NT + tile
  int tile = gt & (NT_ - 1);
  int m    = (gt >> 6) & (M_ - 1);
  int b    = gt >> 8;
  int lane = threadIdx.x;
  int q    = lane & 15;
  int p0   = tile * 16;

  float pv = pre[((size_t)b * P_ + p0 + q) * M_ + m];

  // max over the 16-lane group (both halves identical)
  float mx = pv;
  #pragma unroll
  for (int off = 1; off < 16; off <<= 1) mx = fmaxf(mx, __shfl_xor(mx, off, 32));

  float w = __expf(pv - mx);
  float E = w;
  #pragma unroll
  for (int off = 1; off < 16; off <<= 1) E += __shfl_xor(E, off, 32);

  const float* vt = vals + ((size_t)b * P_ + p0) * D_;
  float s0 = 0.f, s1 = 0.f;
  #pragma unroll
  for (int qq = 0; qq < 16; ++qq) {
    float wq = __shfl(w, qq, 32);
    s0 = fmaf(wq, vt[qq * D_ + lane], s0);
    s1 = fmaf(wq, vt[qq * D_ + lane + 32], s1);
  }

  if (lane == 0) { wsMx[gt] = mx; wsE[gt] = E; }
  wsS[(size_t)gt * D_ + lane]      = s0;
  wsS[(size_t)gt * D_ + lane + 32] = s1;
}

// ---------------------------------------------------------------------------
// Kernel B: per-(b,m) exclusive prefix of the online-softmax state over tiles.
// One wave per (b,m); 64 sequential merge steps (associative softmax merge).
// ---------------------------------------------------------------------------
__global__ void __launch_bounds__(32)
prefix_kernel(const float* __restrict__ wsMx,
              const float* __restrict__ wsE,
              const float* __restrict__ wsS,
              float* __restrict__ wsPM,
              float* __restrict__ wsPE,
              float* __restrict__ wsPS) {
  int g    = blockIdx.x;              // b*M + m
  int lane = threadIdx.x;

  float Mc = -1e30f, Ec = 0.f, sc0 = 0.f, sc1 = 0.f;
  for (int t = 0; t < NT_; ++t) {
    int gt = g * NT_ + t;
    // write exclusive prefix for this tile
    if (lane == 0) { wsPM[gt] = Mc; wsPE[gt] = Ec; }
    wsPS[(size_t)gt * D_ + lane]      = sc0;
    wsPS[(size_t)gt * D_ + lane + 32] = sc1;
    // merge in tile t's local summary
    float mt = wsMx[gt];
    float et = wsE[gt];
    float s0 = wsS[(size_t)gt * D_ + lane];
    float s1 = wsS[(size_t)gt * D_ + lane + 32];
    float Mn = fmaxf(Mc, mt);
    float a  = __expf(Mc - Mn);     // -1e30 carry -> exp underflows to 0
    float bb = __expf(mt - Mn);
    Ec  = fmaf(a, Ec,  bb * et);
    sc0 = fmaf(a, sc0, bb * s0);
    sc1 = fmaf(a, sc1, bb * s1);
    Mc  = Mn;
  }
}

// ---------------------------------------------------------------------------
// Kernel C: per-(b,m,tile) output. Cross-tile part is rank-1 (prefix state);
// diagonal part is tril-exp[16x16] @ V[16x64] done with V_WMMA_F32_16X16X4_F32
// (4 K-steps per 16-wide d-block, f32 exact). One wave per tile.
// ---------------------------------------------------------------------------
__global__ void __launch_bounds__(32)
out_kernel(const float* __restrict__ pre,
           const float* __restrict__ vals,
           const float* __restrict__ mask,
           const float* __restrict__ wsPM,
           const float* __restrict__ wsPE,
           const float* __restrict__ wsPS,
           float* __restrict__ out) {
  int gt   = blockIdx.x;
  int tile = gt & (NT_ - 1);
  int m    = (gt >> 6) & (M_ - 1);
  int b    = gt >> 8;
  int lane = threadIdx.x;
  int hb   = lane >> 4;               // lane half (0 or 1)
  int r    = lane & 15;               // row within tile for this lane
  int p0   = tile * 16;

  float Mp = wsPM[gt];
  float Ep = wsPE[gt];

  float pv = pre[((size_t)b * P_ + p0 + r) * M_ + m];

  // inclusive running max over rows within the 16-lane group
  float sm = pv;
  #pragma unroll
  for (int off = 1; off < 16; off <<= 1) {
    float o = __shfl_up(sm, off, 32);
    if (r >= off) sm = fmaxf(sm, o);
  }
  float rmax  = fmaxf(sm, Mp);        // = cmax[p] of the reference
  float scale = __expf(Mp - rmax);    // rescale factor for the prefix state

  // This lane's 8 entries of the score matrix D(r,c), already in the
  // A-fragment layout of V_WMMA_F32_16X16X4_F32:
  //   k-step t: lanes 0-15 hold (K=4t+0, 4t+1), lanes 16-31 hold (4t+2, 4t+3)
  float dv[8];
  float rowsum = 0.f;
  #pragma unroll
  for (int t = 0; t < 4; ++t) {
    #pragma unroll
    for (int j = 0; j < 2; ++j) {
      int   c  = 4 * t + 2 * hb + j;
      float pc = __shfl(pv, c, 32);
      float d  = (c <= r) ? __expf(pc - rmax) : 0.f;
      dv[2 * t + j] = d;
      rowsum += d;
    }
  }
  rowsum += __shfl_xor(rowsum, 16, 32);          // combine the two lane halves
  float e   = fmaf(scale, Ep, rowsum);           // denominator for row r
  float msk = mask[(size_t)b * P_ + p0 + r];

  // broadcast per-output-row quantities into the C/D layout (row = v + 8*hb)
  float scl8[8], inv8[8], mk8[8];
  #pragma unroll
  for (int v = 0; v < 8; ++v) {
    int row = v + 8 * hb;
    scl8[v] = __shfl(scale, row, 32);
    inv8[v] = 1.0f / __shfl(e, row, 32);
    mk8[v]  = __shfl(msk, row, 32);
  }

  const float* vt = vals + ((size_t)b * P_ + p0) * D_;

  #pragma unroll
  for (int dblk = 0; dblk < 4; ++dblk) {
    int   dcol = dblk * 16 + r;                  // this lane's output column
    float spv  = wsPS[(size_t)gt * D_ + dcol];   // prefix numerator component

    v8f c;
    #pragma unroll
    for (int v = 0; v < 8; ++v) c[v] = scl8[v] * spv;   // rank-1 prefix term

#ifdef USE_WMMA_F32X4
    #pragma unroll
    for (int t = 0; t < 4; ++t) {
      v2f a;  a.x = dv[2 * t];  a.y = dv[2 * t + 1];
      int row0 = 4 * t + 2 * hb;                 // B rows for this lane half
      v2f bf; bf.x = vt[row0 * D_ + dcol];
              bf.y = vt[(row0 + 1) * D_ + dcol];
      c = __builtin_amdgcn_wmma_f32_16x16x4_f32(
              /*neg_a=*/false, a, /*neg_b=*/false, bf,
              /*c_mod=*/(short)0, c, /*reuse_a=*/false, /*reuse_b=*/false);
    }
#else
    // scalar fallback (kept only so the round compiles if builtin is absent)
    #pragma unroll
    for (int v = 0; v < 8; ++v) {
      int   row = v + 8 * hb;
      float rm  = __shfl(rmax, row, 32);
      float acc = c[v];
      #pragma unroll
      for (int q = 0; q < 16; ++q) {
        float pq = __shfl(pv, q, 32);
        float dd = (q <= row) ? __expf(pq - rm) : 0.f;
        acc = fmaf(dd, vt[q * D_ + dcol], acc);
      }
      c[v] = acc;
    }
#endif

    #pragma unroll
    for (int v = 0; v < 8; ++v) {
      int row = v + 8 * hb;
      out[(((size_t)b * P_ + p0 + row) * M_ + m) * D_ + dcol] =
          c[v] * inv8[v] * mk8[v];
    }
  }
}

// ---------------------------------------------------------------------------
extern "C" void kernel_launch(void* const* d_in, const int* in_sizes, int n_in,
                              void* d_out, int out_size, void* d_ws, size_t ws_size,
                              hipStream_t stream) {
  const float* vals = (const float*)d_in[0];   // [B,P,D]
  const float* pre  = (const float*)d_in[1];   // [B,P,M]
  const float* mask = (const float*)d_in[2];   // [B,P,1]
  float*       out  = (float*)d_out;           // [B,P,M,D]

  float* ws   = (float*)d_ws;                  // ~2.1 MB used
  float* wsMx = ws;
  float* wsE  = wsMx + NTOT_;
  float* wsS  = wsE  + NTOT_;
  float* wsPM = wsS  + (size_t)NTOT_ * D_;
  float* wsPE = wsPM + NTOT_;
  float* wsPS = wsPE + NTOT_;

  tile_summary_kernel<<<NTOT_,    32, 0, stream>>>(pre, vals, wsMx, wsE, wsS);
  prefix_kernel      <<<B_ * M_,  32, 0, stream>>>(wsMx, wsE, wsS, wsPM, wsPE, wsPS);
  out_kernel         <<<NTOT_,    32, 0, stream>>>(pre, vals, mask, wsPM, wsPE, wsPS, out);
}